// RangeToBEV_12197707121161
// MI455X (gfx1250) — compile-verified
//
#include <hip/hip_runtime.h>
#include <hip/hip_bf16.h>
#include <math.h>

typedef __attribute__((ext_vector_type(2))) float v2f;
typedef __attribute__((ext_vector_type(8))) float v8f;
typedef unsigned long long u64;

#define BATCH 2
#define CHAN  32
#define NPTS  8192          // H*W = 32*256
#define NXC   440
#define NYC   500
#define NVOXC (NXC*NYC)
#define ROWT  (NPTS/16)     // 512 row tiles per batch
#define COLT  (NPTS/16)     // 512 column tiles
#define PC_X0 0.0f
#define PC_Y0 (-40.0f)
#define PC_Z0 (-3.0f)
#define PC_Z1 1.0f
#define INV_VX (1.0f/0.16f)
#define KEY_INF 0x7f80000000000000ull   // (+inf, idx 0)

// ---------------------------------------------------------------- prep/pack
// near4[t] = (x,y,z, masked ||p||^2)  (invalid near -> +inf norm => d2 = inf)
// far4[t]  = (x,y,z, 0), farn[t] = ||pf||^2
__global__ __launch_bounds__(256) void prep_pack(
    const float* __restrict__ pts, const float* __restrict__ ptsf,
    const int* __restrict__ m,
    float* __restrict__ near4, float* __restrict__ far4,
    float* __restrict__ farn)
{
    int t = blockIdx.x * 256 + threadIdx.x;
    if (t >= BATCH * NPTS) return;
    int b = t / NPTS, n = t - b * NPTS;
    const float* P = pts + (size_t)b * 4 * NPTS;
    float x = P[n], y = P[NPTS + n], z = P[2 * NPTS + n];
    float nn = x * x + y * y + z * z;
    float4 o; o.x = x; o.y = y; o.z = z;
    o.w = (m[t] > 0) ? nn : __builtin_inff();
    ((float4*)near4)[t] = o;
    const float* Pf = ptsf + (size_t)b * 4 * NPTS;
    float xf = Pf[n], yf = Pf[NPTS + n], zf = Pf[2 * NPTS + n];
    float4 of; of.x = xf; of.y = yf; of.z = zf; of.w = 0.0f;
    ((float4*)far4)[t] = of;
    farn[t] = xf * xf + yf * yf + zf * zf;
}

// Branchless sorted-triple insert of a packed (dist,idx) key.
// k0 <= k1 <= k2 ascending (u64 order == dist order since dist >= 0).
// Lowers to v_min_u64 / v_max_u64 on gfx1250.
__device__ __forceinline__ void key_insert(u64 key, u64& k0, u64& k1, u64& k2)
{
    u64 h0 = key > k0 ? key : k0;      // max -> cascades down
    k0     = key < k0 ? key : k0;      // min
    u64 h1 = h0 > k1 ? h0 : k1;
    k1     = h0 < k1 ? h0 : k1;
    k2     = h1 < k2 ? h1 : k2;
}

// ---------------------------------------------------------------- kNN (k=3)
// One wave32 per 16-row tile of far points. Inner loop over 512 column tiles:
// dot tile via V_WMMA_F32_16X16X4_F32 (A = far 16x4, B = near 4x16),
// d2 = nf + nn - 2*dot, packed-key top-3 with uniform (VCC-branch) early-out,
// double-buffered near-tile loads, cross-lane merge in LDS at the end.
__global__ __launch_bounds__(256) void knn3_wmma(
    const float* __restrict__ near4v, const float* __restrict__ far4v,
    const float* __restrict__ farn,
    int* __restrict__ outIdx, float* __restrict__ outW)
{
    __shared__ u64 sk[8][16][16][3];   // [wave][row][colLane][k] = 48 KB

    const int wave = threadIdx.x >> 5;
    const int lane = threadIdx.x & 31;
    const int half = lane >> 4;          // 0: lanes 0-15, 1: lanes 16-31
    const int l16  = lane & 15;
    const int tile = blockIdx.x * 8 + wave;        // over BATCH*ROWT
    const int b = tile / ROWT;
    const int rowBase = (tile - b * ROWT) * 16;

    const float4* np = (const float4*)near4v + (size_t)b * NPTS;
    const float4* fp = (const float4*)far4v  + (size_t)b * NPTS;
    const float*  fn = farn + (size_t)b * NPTS;

    // A-matrix (16x4 f32): lanes 0-15 hold K=0,1 (x,y); lanes 16-31 K=2,3 (z,0)
    float4 fpt = fp[rowBase + l16];
    v2f A;
    A.x = half ? fpt.z : fpt.x;
    A.y = half ? 0.0f  : fpt.y;

    // far norms for my 8 row slots (rows half*8 .. half*8+7)
    float nfr[8];
#pragma unroll
    for (int j = 0; j < 8; ++j) nfr[j] = fn[rowBase + half * 8 + j];

    u64 k0[8], k1[8], k2[8];
#pragma unroll
    for (int j = 0; j < 8; ++j) { k0[j] = k1[j] = k2[j] = KEY_INF; }

    // software-pipelined tile loads: q = current, qn = next (in flight)
    float4 q = np[l16];
    for (int ct = 0; ct < COLT; ++ct) {
        const int colBase = ct * 16;
        const int nc = (ct + 1 < COLT) ? (ct + 1) : (COLT - 1);
        float4 qn = np[nc * 16 + l16];                      // issue next load
        __builtin_prefetch(&np[colBase + 32 + l16], 0, 3);  // tile ct+2 -> near$
        // B-matrix (4x16 f32): VGPR0 lanes0-15 K=0 (x), lanes16-31 K=1 (y);
        //                      VGPR1 lanes0-15 K=2 (z), lanes16-31 K=3 (0)
        v2f Bv;
        Bv.x = half ? q.y  : q.x;
        Bv.y = half ? 0.0f : q.z;
        v8f acc = {0.f, 0.f, 0.f, 0.f, 0.f, 0.f, 0.f, 0.f};
        acc = __builtin_amdgcn_wmma_f32_16x16x4_f32(
            false, A, false, Bv, (short)0, acc, false, false);
        const float nn = q.w;                 // masked near norm for column l16
        const unsigned c = (unsigned)(colBase + l16);   // D layout: N = lane%16
#pragma unroll
        for (int r = 0; r < 8; ++r) {         // D layout: VGPR r -> row half*8+r
            float d2 = fmaxf(fmaf(-2.0f, acc[r], nfr[r] + nn), 0.0f);
            u64 key = ((u64)__float_as_uint(d2) << 32) | c;
            // uniform (VCC) branch: EXEC untouched -> WMMA stays legal
            if (__builtin_amdgcn_ballot_w32(key < k2[r]) != 0u) {
                key_insert(key, k0[r], k1[r], k2[r]);
            }
        }
        q = qn;
    }

#pragma unroll
    for (int r = 0; r < 8; ++r) {
        int row = half * 8 + r;
        sk[wave][row][l16][0] = k0[r];
        sk[wave][row][l16][1] = k1[r];
        sk[wave][row][l16][2] = k2[r];
    }
    __syncthreads();

    if (lane < 16) {
        const int row = l16;
        u64 m0 = KEY_INF, m1 = KEY_INF, m2 = KEY_INF;
        for (int j = 0; j < 16; ++j) {
#pragma unroll
            for (int k = 0; k < 3; ++k)
                key_insert(sk[wave][row][j][k], m0, m1, m2);
        }
        float d0 = __uint_as_float((unsigned)(m0 >> 32));
        float d1 = __uint_as_float((unsigned)(m1 >> 32));
        float d2 = __uint_as_float((unsigned)(m2 >> 32));
        float r0 = 1.0f / (d0 + 1e-8f);
        float r1 = 1.0f / (d1 + 1e-8f);
        float r2 = 1.0f / (d2 + 1e-8f);
        float inv = 1.0f / (r0 + r1 + r2);
        size_t o = ((size_t)b * NPTS + rowBase + row) * 3;
        outIdx[o + 0] = (int)(unsigned)m0;
        outIdx[o + 1] = (int)(unsigned)m1;
        outIdx[o + 2] = (int)(unsigned)m2;
        outW[o + 0] = r0 * inv; outW[o + 1] = r1 * inv; outW[o + 2] = r2 * inv;
    }
}

// ------------------------------------------------------- interpolate+scatter
__device__ __forceinline__ void bev_add(int b, float x, float y, float z,
                                        const float* f,
                                        float* __restrict__ outSum,
                                        float* __restrict__ cnt)
{
    int ix = (int)floorf((x - PC_X0) * INV_VX);
    int iy = (int)floorf((y - PC_Y0) * INV_VX);
    if (ix < 0 || ix >= NXC || iy < 0 || iy >= NYC) return;
    if (!(z >= PC_Z0 && z < PC_Z1)) return;
    int cell = iy * NXC + ix;
    atomicAdd(&cnt[(size_t)b * NVOXC + cell], 1.0f);
    float* o = outSum + (size_t)b * CHAN * NVOXC + cell;   // (B,C,NY,NX)
#pragma unroll
    for (int c = 0; c < CHAN; ++c)
        atomicAdd(&o[(size_t)c * NVOXC], f[c]);
}

__global__ __launch_bounds__(256) void interp_scatter(
    const float* __restrict__ fv, const float* __restrict__ pts,
    const float* __restrict__ ptsf, const int* __restrict__ m,
    const int* __restrict__ mf, const int* __restrict__ kIdx,
    const float* __restrict__ kW, float* __restrict__ outSum,
    float* __restrict__ cnt)
{
    int t = blockIdx.x * 256 + threadIdx.x;
    if (t >= BATCH * NPTS) return;
    int b = t / NPTS, n = t - b * NPTS;
    const float* F = fv + (size_t)b * CHAN * NPTS;
    if (m[t] > 0) {
        const float* P = pts + (size_t)b * 4 * NPTS;
        float f[CHAN];
#pragma unroll
        for (int c = 0; c < CHAN; ++c) f[c] = F[(size_t)c * NPTS + n];
        bev_add(b, P[n], P[NPTS + n], P[2 * NPTS + n], f, outSum, cnt);
    }
    if (mf[t] > 0) {
        const float* Pf = ptsf + (size_t)b * 4 * NPTS;
        size_t o = (size_t)t * 3;
        int a0 = kIdx[o + 0]; a0 = (a0 < 0 || a0 >= NPTS) ? 0 : a0;
        int a1 = kIdx[o + 1]; a1 = (a1 < 0 || a1 >= NPTS) ? 0 : a1;
        int a2 = kIdx[o + 2]; a2 = (a2 < 0 || a2 >= NPTS) ? 0 : a2;
        float w0 = kW[o + 0], w1 = kW[o + 1], w2 = kW[o + 2];
        float f[CHAN];
#pragma unroll
        for (int c = 0; c < CHAN; ++c) {
            const float* Fc = F + (size_t)c * NPTS;
            f[c] = w0 * Fc[a0] + w1 * Fc[a1] + w2 * Fc[a2];
        }
        bev_add(b, Pf[n], Pf[NPTS + n], Pf[2 * NPTS + n], f, outSum, cnt);
    }
}

// ---------------------------------------------------------------- normalize
__global__ __launch_bounds__(256) void bev_norm(float* __restrict__ out,
                                                const float* __restrict__ cnt)
{
    size_t t = (size_t)blockIdx.x * 256 + threadIdx.x;
    if (t >= (size_t)BATCH * CHAN * NVOXC) return;
    size_t cell = t % NVOXC;
    size_t b = (t / NVOXC) / CHAN;
    float c = cnt[b * NVOXC + cell];
    out[t] = out[t] / fmaxf(c, 1.0f);
}

// ---------------------------------------------------------------- launcher
extern "C" void kernel_launch(void* const* d_in, const int* in_sizes, int n_in,
                              void* d_out, int out_size, void* d_ws, size_t ws_size,
                              hipStream_t stream)
{
    const float* fv   = (const float*)d_in[0];
    const float* pts  = (const float*)d_in[1];
    const float* ptsf = (const float*)d_in[2];
    const int*   m    = (const int*)d_in[3];
    const int*   mf   = (const int*)d_in[4];
    float* out = (float*)d_out;

    char* ws = (char*)d_ws;
    float* near4 = (float*)ws; ws += (size_t)BATCH * NPTS * 4 * sizeof(float);
    float* far4  = (float*)ws; ws += (size_t)BATCH * NPTS * 4 * sizeof(float);
    float* farn  = (float*)ws; ws += (size_t)BATCH * NPTS * sizeof(float);
    int*   kIdx  = (int*)ws;   ws += (size_t)BATCH * NPTS * 3 * sizeof(int);
    float* kW    = (float*)ws; ws += (size_t)BATCH * NPTS * 3 * sizeof(float);
    float* cnt   = (float*)ws;

    hipMemsetAsync(d_out, 0, (size_t)out_size * sizeof(float), stream);
    hipMemsetAsync(cnt, 0, (size_t)BATCH * NVOXC * sizeof(float), stream);

    prep_pack<<<(BATCH * NPTS + 255) / 256, 256, 0, stream>>>(
        pts, ptsf, m, near4, far4, farn);
    knn3_wmma<<<(BATCH * ROWT) / 8, 256, 0, stream>>>(
        near4, far4, farn, kIdx, kW);
    interp_scatter<<<(BATCH * NPTS + 255) / 256, 256, 0, stream>>>(
        fv, pts, ptsf, m, mf, kIdx, kW, out, cnt);
    bev_norm<<<(unsigned)(((size_t)BATCH * CHAN * NVOXC + 255) / 256), 256, 0, stream>>>(
        out, cnt);
}